// StreamingAttentionSink_48395691491451
// MI455X (gfx1250) — compile-verified
//
#include <hip/hip_runtime.h>
#include <hip/hip_bf16.h>

#define B_    16
#define H_    8
#define D_    128
#define BS_   16
#define L_    4096
#define MAXB_ 384
#define HID_  1024
#define T_    (L_ + 1)          // 4097 keys total per (b,h)
#define NWIN  (L_ / BS_ - 2)    // 254 window blocks
#define CH_   32                // keys staged per chunk
#define NCH_  (L_ / CH_)        // 128 chunks cover the 4096 cached keys

typedef __attribute__((ext_vector_type(2))) float v2f;
typedef __attribute__((ext_vector_type(8))) float v8f;

// ---- gfx1250 async global->LDS copy (ASYNCcnt-tracked) ----
// Each lane moves 16B from its global address to its LDS address.
// LDS aperture uses addr[31:0] of the flat shared pointer (ISA 10.2).
__device__ __forceinline__ void async_copy_b128(const float* __restrict__ g,
                                                float* __restrict__ l) {
    unsigned int lds_addr = (unsigned int)(unsigned long long)l;
    asm volatile("global_load_async_to_lds_b128 %0, %1, off"
                 :: "v"(lds_addr), "v"(g) : "memory");
}
#define ASYNC_WAIT(n) asm volatile("s_wait_asynccnt " #n ::: "memory")

// Resolve the global row pointer for logical key index t (0..4096).
// cur = pointer to the current-token k/v head slice (used when t == L_).
__device__ __forceinline__ const float* kv_row(const float* __restrict__ cache,
                                               const float* __restrict__ cur,
                                               const int* __restrict__ bt_b,
                                               int m, int t, int h) {
    if (t >= L_) return cur;
    int phys, r;
    if (t < BS_) {                       // sink block
        phys = bt_b[0];
        r = t;
    } else if (t < L_ - BS_) {           // sliding window blocks
        int u = t - BS_;
        phys = bt_b[m - NWIN + (u >> 4)];
        r = u & (BS_ - 1);
    } else {                             // current (partial) block
        phys = bt_b[m];
        r = t - (L_ - BS_);
    }
    return cache + (((size_t)phys * BS_ + r) * H_ + h) * D_;
}

__device__ __forceinline__ void key_pos_mask(int t, int rem, float* pos, bool* ok) {
    if (t < BS_)             { *pos = (float)t;                           *ok = true; }
    else if (t < L_ - BS_)   { *pos = (float)(2*BS_ - 1 - rem + (t-BS_)); *ok = true; }
    else if (t < L_)         { int u = t - (L_ - BS_);
                               *pos = (float)(L_ - BS_ - 1 - rem + u);    *ok = (u < rem); }
    else                     { *pos = (float)(L_ - 1);                    *ok = true; }
}

__global__ __launch_bounds__(128)
void attn_kernel(const float* __restrict__ q, const float* __restrict__ k,
                 const float* __restrict__ v,
                 const float* __restrict__ key_cache,
                 const float* __restrict__ value_cache,
                 const int* __restrict__ block_tables,
                 const int* __restrict__ seq_lens,
                 float* __restrict__ attn_out) {
    const int b = blockIdx.x / H_;
    const int h = blockIdx.x % H_;
    const int tid  = threadIdx.x;
    const int lane = tid & 31;
    const int wave = tid >> 5;

    __shared__ float s_kv[2][CH_][D_];      // 32 KB double-buffered staging
    __shared__ float s_scores[T_ + 15];
    __shared__ float s_q1[D_ / 2];
    __shared__ float s_q2[D_ / 2];
    __shared__ float s_red[8];
    __shared__ float s_vpart[4][D_];

    const int seq = seq_lens[b];
    const int num_past = seq - 1;
    const int m   = num_past / BS_;
    const int rem = num_past % BS_;
    const int* bt_b = block_tables + b * MAXB_;
    const float scale = 0.08838834764831845f;   // 1/sqrt(128)

    // ---- RoPE the query at pos L-1 (once) ----
    if (tid < D_ / 2) {
        const int j = tid;
        const float invf = __powf(10000.0f, -(float)j / 64.0f);
        float sn, cs;
        __sincosf((float)(L_ - 1) * invf, &sn, &cs);
        const float q1 = q[b * HID_ + h * D_ + j];
        const float q2 = q[b * HID_ + h * D_ + j + 64];
        s_q1[j] = q1 * cs - q2 * sn;
        s_q2[j] = q2 * cs + q1 * sn;
    }
    __syncthreads();

    // Per-lane constants: this lane owns RoPE pair indices j0=2*lane, j0+1
    const int j0 = 2 * lane;
    const float if0 = __powf(10000.0f, -(float)j0 / 64.0f);
    const float if1 = __powf(10000.0f, -(float)(j0 + 1) / 64.0f);
    const float qa0 = s_q1[j0], qb0 = s_q2[j0];
    const float qa1 = s_q1[j0 + 1], qb1 = s_q2[j0 + 1];

    const float* kcur = k + b * HID_ + h * D_;
    const float* vcur = v + b * HID_ + h * D_;

    // Stage CH_ rows of chunk c into s_kv[pp]: wave w moves rows {4i+w},
    // lane moves 16B -> one async b128 per row per wave, 8 per chunk per wave.
    #define STAGE_CHUNK(cache, cur, c, pp)                                    \
        _Pragma("unroll")                                                     \
        for (int i = 0; i < CH_ / 4; ++i) {                                   \
            const int r = i * 4 + wave;                                       \
            const float* src = kv_row(cache, cur, bt_b, m, (c) * CH_ + r, h); \
            async_copy_b128(src + lane * 4, &s_kv[pp][r][lane * 4]);          \
        }

    // ========== Phase 1: masked RoPE'd scores into LDS ==========
    STAGE_CHUNK(key_cache, kcur, 0, 0);
    int pp = 0;
    for (int c = 0; c < NCH_; ++c) {
        if (c + 1 < NCH_) {
            STAGE_CHUNK(key_cache, kcur, c + 1, pp ^ 1);
            ASYNC_WAIT(8);        // chunk c complete (in-order), c+1 in flight
        } else {
            ASYNC_WAIT(0);
        }
        __syncthreads();
        for (int r = wave; r < CH_; r += 4) {
            const int t = c * CH_ + r;
            float pos; bool ok;
            key_pos_mask(t, rem, &pos, &ok);
            const float* kr = &s_kv[pp][r][0];
            const float2 k1 = *(const float2*)(kr + j0);
            const float2 k2 = *(const float2*)(kr + 64 + j0);
            float s0, c0, s1, c1;
            __sincosf(pos * if0, &s0, &c0);
            __sincosf(pos * if1, &s1, &c1);
            // q_rot . K_rot == k1*(q1r*c + q2r*s) + k2*(q2r*c - q1r*s)
            float acc = k1.x * (qa0 * c0 + qb0 * s0) + k2.x * (qb0 * c0 - qa0 * s0)
                      + k1.y * (qa1 * c1 + qb1 * s1) + k2.y * (qb1 * c1 - qa1 * s1);
            #pragma unroll
            for (int off = 16; off > 0; off >>= 1)
                acc += __shfl_xor(acc, off, 32);
            if (lane == 0)
                s_scores[t] = ok ? acc * scale : -INFINITY;
        }
        __syncthreads();          // chunk consumed; its buffer may be re-staged
        pp ^= 1;
    }
    // Current token (t = L_): wave 0, direct from k
    if (wave == 0) {
        const float2 k1 = *(const float2*)(kcur + j0);
        const float2 k2 = *(const float2*)(kcur + 64 + j0);
        float s0, c0, s1, c1;
        __sincosf((float)(L_ - 1) * if0, &s0, &c0);
        __sincosf((float)(L_ - 1) * if1, &s1, &c1);
        float acc = k1.x * (qa0 * c0 + qb0 * s0) + k2.x * (qb0 * c0 - qa0 * s0)
                  + k1.y * (qa1 * c1 + qb1 * s1) + k2.y * (qb1 * c1 - qa1 * s1);
        #pragma unroll
        for (int off = 16; off > 0; off >>= 1)
            acc += __shfl_xor(acc, off, 32);
        if (lane == 0) s_scores[L_] = acc * scale;
    }
    __syncthreads();

    // ========== Phase 2: softmax over T_ ==========
    float lmax = -INFINITY;
    for (int t = tid; t < T_; t += 128) lmax = fmaxf(lmax, s_scores[t]);
    #pragma unroll
    for (int off = 16; off > 0; off >>= 1)
        lmax = fmaxf(lmax, __shfl_xor(lmax, off, 32));
    if (lane == 0) s_red[wave] = lmax;
    __syncthreads();
    const float mx = fmaxf(fmaxf(s_red[0], s_red[1]), fmaxf(s_red[2], s_red[3]));

    float lsum = 0.0f;
    for (int t = tid; t < T_; t += 128) {
        const float p = __expf(s_scores[t] - mx);   // exp(-inf)=0 handles mask
        s_scores[t] = p;
        lsum += p;
    }
    #pragma unroll
    for (int off = 16; off > 0; off >>= 1)
        lsum += __shfl_xor(lsum, off, 32);
    if (lane == 0) s_red[4 + wave] = lsum;
    __syncthreads();
    const float inv_sum = 1.0f / (s_red[4] + s_red[5] + s_red[6] + s_red[7]);

    // ========== Phase 3: out = sum_t p_t * V[t] (same async pipeline) ==========
    float4 acc4 = make_float4(0.f, 0.f, 0.f, 0.f);
    STAGE_CHUNK(value_cache, vcur, 0, 0);
    pp = 0;
    for (int c = 0; c < NCH_; ++c) {
        if (c + 1 < NCH_) {
            STAGE_CHUNK(value_cache, vcur, c + 1, pp ^ 1);
            ASYNC_WAIT(8);
        } else {
            ASYNC_WAIT(0);
        }
        __syncthreads();
        for (int r = wave; r < CH_; r += 4) {
            const float p = s_scores[c * CH_ + r];
            const float4 vv = *(const float4*)&s_kv[pp][r][4 * lane];
            acc4.x += p * vv.x; acc4.y += p * vv.y;
            acc4.z += p * vv.z; acc4.w += p * vv.w;
        }
        __syncthreads();
        pp ^= 1;
    }
    *(float4*)&s_vpart[wave][4 * lane] = acc4;
    __syncthreads();

    if (tid < D_) {
        float o = s_vpart[0][tid] + s_vpart[1][tid] +
                  s_vpart[2][tid] + s_vpart[3][tid];
        o += s_scores[L_] * vcur[tid];              // current token's V
        attn_out[b * HID_ + h * D_ + tid] = o * inv_sum;
    }
    #undef STAGE_CHUNK
}

// out(16x1024) = A(16x1024) @ W(1024x1024), full fp32 via V_WMMA_F32_16X16X4_F32.
// A layout (16x4 f32): lanes 0-15 = rows, VGPR0 holds K=0 (lo half)/K=2 (hi half),
// VGPR1 holds K=1/K=3. B/C/D: rows striped across lanes within a VGPR.
__global__ __launch_bounds__(256)
void proj_kernel(const float* __restrict__ A, const float* __restrict__ W,
                 float* __restrict__ out) {
    const int lane = threadIdx.x & 31;
    const int wave = threadIdx.x >> 5;
    const int half = lane >> 4;         // 0: lanes 0-15, 1: lanes 16-31
    const int ln   = lane & 15;
    const int n0   = (blockIdx.x * 8 + wave) * 16;

    v8f c = {0.f, 0.f, 0.f, 0.f, 0.f, 0.f, 0.f, 0.f};
    for (int kk = 0; kk < HID_; kk += 4) {
        v2f a, bm;
        const float* ap = A + ln * HID_ + kk + 2 * half;
        a.x = ap[0];                 // K = kk + 2*half
        a.y = ap[1];                 // K = kk + 2*half + 1
        const float* bp = W + (size_t)(kk + 2 * half) * HID_ + n0 + ln;
        bm.x = bp[0];                // row kk + 2*half
        bm.y = bp[HID_];             // row kk + 2*half + 1
        c = __builtin_amdgcn_wmma_f32_16x16x4_f32(false, a, false, bm,
                                                  (short)0, c, false, false);
    }
    #pragma unroll
    for (int r = 0; r < 8; ++r)
        out[(size_t)(r + 8 * half) * HID_ + n0 + ln] = c[r];
}

extern "C" void kernel_launch(void* const* d_in, const int* in_sizes, int n_in,
                              void* d_out, int out_size, void* d_ws, size_t ws_size,
                              hipStream_t stream) {
    const float* q           = (const float*)d_in[0];
    const float* k           = (const float*)d_in[1];
    const float* v           = (const float*)d_in[2];
    // d_in[3] = positions (unused; recomputed from seq_lens)
    const float* key_cache   = (const float*)d_in[4];
    const float* value_cache = (const float*)d_in[5];
    const int*   block_tables= (const int*)d_in[6];
    const int*   seq_lens    = (const int*)d_in[7];
    const float* W_o         = (const float*)d_in[8];
    float* out               = (float*)d_out;
    float* attn_ws           = (float*)d_ws;            // B_*HID_ floats

    attn_kernel<<<dim3(B_ * H_), dim3(128), 0, stream>>>(
        q, k, v, key_cache, value_cache, block_tables, seq_lens, attn_ws);
    proj_kernel<<<dim3(HID_ / 16 / 8), dim3(256), 0, stream>>>(attn_ws, W_o, out);
}